// IterativeQuadInterp3d_2353642078939
// MI455X (gfx1250) — compile-verified
//
#include <hip/hip_runtime.h>
#include <stdint.h>

#ifndef __has_builtin
#define __has_builtin(x) 0
#endif

namespace {
constexpr int kB   = 2;      // B*C volumes
constexpr int kD   = 6;
constexpr int kH   = 320;
constexpr int kW   = 320;
constexpr int kHW  = kH * kW;
constexpr int kDHW = kD * kHW;

constexpr int kTile = 16;
constexpr int kHalo = 5;                   // 4 center moves + patch radius 1
constexpr int kP    = kTile + 2 * kHalo;   // 26  (LDS pitch)
constexpr int kPP   = kP * kP;             // 676
constexpr int kLds  = kD * kPP;            // 4056 floats = 16.2 KB
constexpr int kThreads = kTile * kTile;    // 256 = 8 waves (wave32)

constexpr int   kIters    = 5;
constexpr float kBonus    = 10.0f;
constexpr float kMaxShift = 0.6f;
constexpr float kEps      = 1e-7f;
} // namespace

__device__ __forceinline__ int iclamp(int v, int lo, int hi) {
    return v < lo ? lo : (v > hi ? hi : v);
}

__global__ __launch_bounds__(kThreads)
void iqi3d_kernel(const float* __restrict__ x,
                  float* __restrict__ coords,
                  float* __restrict__ yout)
{
    __shared__ float tile[kLds];

    const int bc = blockIdx.z;
    const int x0 = blockIdx.x * kTile;
    const int y0 = blockIdx.y * kTile;
    const int ox = x0 - kHalo;
    const int oy = y0 - kHalo;
    const float* __restrict__ vol = x + (size_t)bc * kDHW;

    const int tid = threadIdx.y * kTile + threadIdx.x;

    // ---- Stage replicate-clamped tile into LDS via CDNA5 async copies ----
    // Clamping the global coordinate replicates volume edges into the halo,
    // which exactly reproduces the reference's 'edge' padding for NMS.
    for (int e = tid; e < kLds; e += kThreads) {
        const int z  = e / kPP;
        const int r  = e - z * kPP;
        const int yy = r / kP;
        const int xx = r - yy * kP;
        const int gy = iclamp(oy + yy, 0, kH - 1);
        const int gx = iclamp(ox + xx, 0, kW - 1);
        const uint64_t gaddr = (uint64_t)(size_t)(vol + ((z * kH + gy) * kW + gx));
        const uint32_t laddr = (uint32_t)(size_t)(&tile[e]);
        asm volatile("global_load_async_to_lds_b32 %0, %1, off"
                     :: "v"(laddr), "v"(gaddr)
                     : "memory");
    }
#if __has_builtin(__builtin_amdgcn_s_wait_asynccnt)
    __builtin_amdgcn_s_wait_asynccnt(0);
#else
    asm volatile("s_wait_asynccnt 0" ::: "memory");
#endif
    __syncthreads();

    const int gx  = x0 + threadIdx.x;
    const int gy  = y0 + threadIdx.y;
    const int lyc = threadIdx.y + kHalo;
    const int lxc = threadIdx.x + kHalo;

    float* __restrict__ cS = coords + (size_t)bc * 3 * kDHW;  // [coord_s, coord_x, coord_y]
    float* __restrict__ cX = cS + kDHW;
    float* __restrict__ cY = cX + kDHW;
    float* __restrict__ yo = yout + (size_t)bc * kDHW;
    const size_t spBase = (size_t)gy * kW + gx;

    // ---- Per-plane 3x3 stats for the whole depth column (registers) ----
    // a8 = max of 8 ring neighbors, a9 = max(a8, center), ctr = center.
    // m(z) = max(a9[clamp(z-1)], a8[z], a9[clamp(z+1)], 0) reproduces the
    // 26-neighbor replicate-padded max exactly (incl. forced-false z borders).
    float a8[kD], a9[kD], ctr[kD];
    #pragma unroll
    for (int z = 0; z < kD; ++z) {
        const float* r0 = &tile[(z * kP + (lyc - 1)) * kP + (lxc - 1)];
        const float n00 = r0[0],        n01 = r0[1],          n02 = r0[2];
        const float n10 = r0[kP],       cc  = r0[kP + 1],     n12 = r0[kP + 2];
        const float n20 = r0[2 * kP],   n21 = r0[2 * kP + 1], n22 = r0[2 * kP + 2];
        const float m8 = fmaxf(fmaxf(fmaxf(n00, n01), fmaxf(n02, n10)),
                               fmaxf(fmaxf(n12, n20), fmaxf(n21, n22)));
        a8[z]  = m8;
        a9[z]  = fmaxf(m8, cc);
        ctr[z] = cc;
    }

    #pragma unroll
    for (int z = 0; z < kD; ++z) {
        const int zm = (z > 0) ? z - 1 : 0;
        const int zp = (z < kD - 1) ? z + 1 : kD - 1;
        const float c = ctr[z];
        const float m = fmaxf(fmaxf(a9[zm], a8[z]), fmaxf(a9[zp], 0.0f));

        float o_cs = (float)z;
        float o_cx = (float)gx;
        float o_cy = (float)gy;
        float o_y  = c;

        // Refinement only matters where kp = mask & valid; mask==false voxels
        // emit grid coords + unchanged score, so skip the solve entirely.
        if (c > m) {
            int dc = z, hc = gy, wc = gx;
            bool valid = true;
            float sx = 0.f, sy = 0.f, ss = 0.f, gdot = 0.f;

            #pragma unroll 1
            for (int it = 0; it < kIters; ++it) {
                const int dsb = iclamp(dc, 1, kD - 2);
                const int hsb = iclamp(hc, 1, kH - 2);
                const int wsb = iclamp(wc, 1, kW - 2);
                // base at (d-1, h-1, w-1): all 19 used taps are +const offsets
                const float* p =
                    &tile[((dsb - 1) * kP + (hsb - 1 - oy)) * kP + (wsb - 1 - ox)];

                const float smm  = p[kP + 0];
                const float sm_  = p[kP + 1];
                const float smp  = p[kP + 2];
                const float tmm  = p[1];
                const float tmp_ = p[2 * kP + 1];
                const float hmm  = p[kPP + 0];
                const float ym   = p[kPP + 1];
                const float hmp  = p[kPP + 2];
                const float xm   = p[kPP + kP + 0];
                const float c000 = p[kPP + kP + 1];
                const float xp   = p[kPP + kP + 2];
                const float hpm  = p[kPP + 2 * kP + 0];
                const float yp   = p[kPP + 2 * kP + 1];
                const float hpp  = p[kPP + 2 * kP + 2];
                const float tpm  = p[2 * kPP + 1];
                const float spm  = p[2 * kPP + kP + 0];
                const float sp_  = p[2 * kPP + kP + 1];
                const float spp  = p[2 * kPP + kP + 2];
                const float tpp  = p[2 * kPP + 2 * kP + 1];

                const float gx_ = 0.5f * (xp - xm);
                const float gy_ = 0.5f * (yp - ym);
                const float gs_ = 0.5f * (sp_ - sm_);
                const float dxx = xp - 2.0f * c000 + xm;
                const float dyy = yp - 2.0f * c000 + ym;
                const float dss = sp_ - 2.0f * c000 + sm_;
                const float dxy = 0.25f * (hpp - hpm - hmp + hmm);
                const float dxs = 0.25f * (spp - spm - smp + smm);
                const float dys = 0.25f * (tpp - tpm - tmp_ + tmm);

                const float cf00 = dyy * dss - dys * dys;
                const float cf01 = dxy * dss - dys * dxs;
                const float cf02 = dxy * dys - dyy * dxs;
                const float det  = dxx * cf00 - dxy * cf01 + dxs * cf02;
                if (!(fabsf(det) > kEps)) { valid = false; break; }

                const float r0 = -gx_, r1 = -gy_, r2 = -gs_;
                const float t0 = r1 * dss - dys * r2;
                const float t1 = r1 * dys - dyy * r2;
                const float t2 = dxy * r2 - r1 * dxs;
                const float t3 = dyy * r2 - r1 * dys;
                sx = (r0 * cf00 - dxy * t0 + dxs * t1) / det;
                sy = (dxx * t0 - r0 * cf01 + dxs * t2) / det;
                ss = (dxx * t3 - dxy * t2 + r0 * cf02) / det;
                gdot = gx_ * sx + gy_ * sy + gs_ * ss;

                // integer center moves: x, then y, then d (each bound check
                // gates the next move, exactly like the reference)
                const int mvx = ((sx >  kMaxShift) ? 1 : 0) - ((sx < -kMaxShift) ? 1 : 0);
                const int nw = wc + mvx;
                if (nw < 1 || nw > kW - 2) { valid = false; break; }
                wc = nw;
                const int mvy = ((sy >  kMaxShift) ? 1 : 0) - ((sy < -kMaxShift) ? 1 : 0);
                const int nh = hc + mvy;
                if (nh < 1 || nh > kH - 2) { valid = false; break; }
                hc = nh;
                const int mvd = ((ss >  kMaxShift) ? 1 : 0) - ((ss < -kMaxShift) ? 1 : 0);
                const int nd = dc + mvd;
                if (nd < 1 || nd > kD - 2) { valid = false; break; }
                dc = nd;

                // Fixed point: no movement => every remaining iteration
                // recomputes the identical patch/solve/checks. Break is
                // bit-identical to running out the loop.
                if ((mvx | mvy | mvd) == 0) break;
            }

            if (valid && fabsf(sx) <= 1.5f && fabsf(sy) <= 1.5f && fabsf(ss) <= 1.5f) {
                o_cs = (float)dc + ss;
                o_cx = (float)wc + sx;
                o_cy = (float)hc + sy;
                o_y  = c + 0.5f * gdot + kBonus;
            }
        }

        const size_t idx = (size_t)z * kHW + spBase;
        cS[idx] = o_cs;
        cX[idx] = o_cx;
        cY[idx] = o_cy;
        yo[idx] = o_y;
    }
}

extern "C" void kernel_launch(void* const* d_in, const int* in_sizes, int n_in,
                              void* d_out, int out_size, void* d_ws, size_t ws_size,
                              hipStream_t stream) {
    (void)in_sizes; (void)n_in; (void)out_size; (void)d_ws; (void)ws_size;
    const float* x = (const float*)d_in[0];
    float* out    = (float*)d_out;
    float* coords = out;                              // B*3*D*H*W
    float* y      = out + (size_t)kB * 3 * kDHW;      // B*D*H*W

    dim3 grid(kW / kTile, kH / kTile, kB);
    dim3 block(kTile, kTile, 1);
    hipLaunchKernelGGL(iqi3d_kernel, grid, block, 0, stream, x, coords, y);
}